// ICELUT_15968688407028
// MI455X (gfx1250) — compile-verified
//
#include <hip/hip_runtime.h>
#include <hip/hip_bf16.h>

// ---------------------------------------------------------------------------
// ICELUT pipeline for gfx1250 (MI455X)
//   k1: feature gather + per-block partial sums
//   k2: final pooled mean -> quantize -> lut_cat select -> weights[10]
//   kp: zero-pad s_layers / luts / w_layers into WMMA-friendly buffers
//   k3: Tpad = luts @ w_layers          [V_WMMA_F32_16X16X4_F32, 3 k-steps]
//   k4: cubes2 = s_layers @ T.view(5,~) [V_WMMA_F32_16X16X4_F32, branch-free]
//   k5: d3lut(3x33^3) = sum_n w[n] * permuted(cubes2)  -> d_out tail
//   k6: trilinear interp (paired b64 gathers), +img_org -> d_out head
// ---------------------------------------------------------------------------

#define DIMV   33
#define NUMV   10
#define NFEAT  10
#define HW_S   (256 * 256)          // 65536
#define H_ORG  2160
#define W_ORG  3840
#define HW_ORG (H_ORG * W_ORG)      // 8294400
#define NCOL   32670                // NUM*3*DIM*DIM
#define NCOLP  32672                // padded stride (multiple of 16)
#define CUBE3  35937                // 33^3
#define PLANE2 1089                 // 33^2

// workspace layout (in floats)
#define WS_PARTIALS 0               // 256 blocks * 16           = 4096
#define WS_WEIGHTS  4096            // 10 (pad 16)
#define WS_SPAD     4112            // 48 x 5                    = 240
#define WS_LPAD     4352            // 160 x 12                  = 1920
#define WS_WPAD     6272            // 12 x 1104                 = 13248
#define WS_TPAD     19520           // 5 x 32672                 = 163360
#define WS_CUBES2   182880          // 48 x 32672                = 1568256
// total: 1,751,136 floats = ~7.0 MB

typedef float v2f __attribute__((ext_vector_type(2)));
typedef float v8f __attribute__((ext_vector_type(8)));

struct fpair { float lo, hi; };     // align(4), size 8 -> mergeable b64 load

__device__ __forceinline__ int iclamp(int v, int lo, int hi) {
    return v < lo ? lo : (v > hi ? hi : v);
}

// ---------------- k1: feature gather + block-level partial sums -------------
__global__ void k_feat_partial(const int* __restrict__ msb,
                               const int* __restrict__ lsb,
                               const float* __restrict__ fmsb,
                               const float* __restrict__ flsb,
                               float* __restrict__ partials) {
    __shared__ float sred[256];
    const int tid = threadIdx.x;
    const int p   = blockIdx.x * 256 + tid;   // p < 65536

    const int im = msb[p] * 4096 + msb[HW_S + p] * 256 + msb[2 * HW_S + p] * 16;
    const int il = lsb[p] * 4096 + lsb[HW_S + p] * 256 + lsb[2 * HW_S + p] * 16;
    const float* rm = fmsb + (size_t)im * NFEAT;
    const float* rl = flsb + (size_t)il * NFEAT;

    float f[NFEAT];
#pragma unroll
    for (int j = 0; j < NFEAT; ++j) f[j] = rm[j] + rl[j];

    for (int j = 0; j < NFEAT; ++j) {
        sred[tid] = f[j];
        __syncthreads();
        for (int s = 128; s > 0; s >>= 1) {
            if (tid < s) sred[tid] += sred[tid + s];
            __syncthreads();
        }
        if (tid == 0) partials[blockIdx.x * 16 + j] = sred[0];
        __syncthreads();
    }
}

// ---------------- k2: final reduce + quantize + weight selection ------------
__global__ void k_weights(const float* __restrict__ partials,
                          const float* __restrict__ lut_cat,
                          float* __restrict__ weights) {
    __shared__ float sred[256];
    __shared__ float pooled[NFEAT];
    const int tid = threadIdx.x;

    for (int j = 0; j < NFEAT; ++j) {
        sred[tid] = partials[tid * 16 + j];
        __syncthreads();
        for (int s = 128; s > 0; s >>= 1) {
            if (tid < s) sred[tid] += sred[tid + s];
            __syncthreads();
        }
        if (tid == 0) pooled[j] = sred[0] * (1.0f / 65536.0f);
        __syncthreads();
    }

    if (tid == 0) {
        int mid[NFEAT];
#pragma unroll
        for (int j = 0; j < NFEAT; ++j) {
            float p2 = rintf(pooled[j] * 2.0f);    // round-half-to-even
            p2 = fminf(fmaxf(p2, -32.0f), 31.0f);
            mid[j] = (int)p2 + 32;                 // [0,63]
        }
        float w[NFEAT];
#pragma unroll
        for (int j = 0; j < NFEAT; ++j) w[j] = 0.0f;
        for (int i = 0; i < 5; ++i) {
            const int index = mid[2 * i] * 64 + mid[2 * i + 1];
            const float* sel = lut_cat + ((size_t)i * 4096 + index) * NFEAT;
#pragma unroll
            for (int j = 0; j < NFEAT; ++j) w[j] += (sel[j] - 32.0f) * 0.25f;
        }
#pragma unroll
        for (int j = 0; j < NFEAT; ++j) weights[j] = w[j];
    }
}

// ---------------- kp: zero-pad operands for branch-free WMMA ----------------
__global__ void k_pad(const float* __restrict__ s_layers,   // 33x5
                      const float* __restrict__ luts,       // 150x10
                      const float* __restrict__ wl,         // 10x1089
                      float* __restrict__ Spad,             // 48x5
                      float* __restrict__ Lpad,             // 160x12
                      float* __restrict__ Wpad) {           // 12x1104
    const int i = blockIdx.x * blockDim.x + threadIdx.x;
    if (i < 240) {
        const int r = i / 5;
        Spad[i] = (r < 33) ? s_layers[i] : 0.0f;
    }
    if (i < 1920) {
        const int r = i / 12, k = i % 12;
        Lpad[i] = (r < 150 && k < 10) ? luts[r * 10 + k] : 0.0f;
    }
    if (i < 13248) {
        const int k = i / 1104, n = i % 1104;
        Wpad[i] = (k < 10 && n < 1089) ? wl[k * PLANE2 + n] : 0.0f;
    }
}

// ---------------- k3: Tpad = luts @ w_layers  via WMMA ----------------------
// M=160 (10 tiles), N=1104 (69 tiles), K=12 (3 k-steps of 16x16x4).
// A (16x4 f32): lanes 0-15 -> M rows; vgpr {x,y} = K {kb,kb+1}, kb=(lane>>4)*2
// B (4x16 f32): lanes 0-15 -> N cols; vgpr {x,y} = K {kb,kb+1}
// C/D: vgpr v, lane l -> M = v + (l>>4)*8, N = l&15
// Output scattered into the padded (5 x 32672) view of T.
__global__ void k_lutsw_wmma(const float* __restrict__ Lpad,
                             const float* __restrict__ Wpad,
                             float* __restrict__ Tpad) {
    const int gtid = blockIdx.x * blockDim.x + threadIdx.x;
    const int wave = gtid >> 5;
    const int lane = gtid & 31;
    const int ntiles = 69;
    if (wave >= 10 * ntiles) return;

    const int mt = wave / ntiles;
    const int nt = wave % ntiles;
    const int m0 = mt * 16, n0 = nt * 16;
    const int lm = lane & 15;
    const int lh = lane >> 4;
    const int kb = lh * 2;
    const int m  = m0 + lm;
    const int n  = n0 + lm;

    v8f acc = {};
#pragma unroll
    for (int ks = 0; ks < 3; ++ks) {
        const int k0 = ks * 4 + kb;
        v2f a, b;
        a.x = Lpad[m * 12 + k0];
        a.y = Lpad[m * 12 + k0 + 1];
        b.x = Wpad[k0 * 1104 + n];
        b.y = Wpad[(k0 + 1) * 1104 + n];
        acc = __builtin_amdgcn_wmma_f32_16x16x4_f32(false, a, false, b,
                                                    (short)0, acc, false, false);
    }
#pragma unroll
    for (int v = 0; v < 8; ++v) {
        const int r = m0 + v + lh * 8;
        if (r < 150 && n < PLANE2) {
            const int s = r / 30, mm = r % 30;
            Tpad[s * NCOLP + mm * PLANE2 + n] = acc[v];
        }
    }
}

// ---------------- k4: cubes2 = s_layers @ T.view(5, NCOLP)  via WMMA --------
// Branch-free: padded A (48x5), padded B rows (stride NCOLP), padded C
// (48 x NCOLP). K=5 = one K=4 step + one K=1 step (zero-filled slots via
// v_cndmask, no exec divergence).
__global__ void k_cubes_wmma(const float* __restrict__ Spad,  // 48x5
                             const float* __restrict__ Tpad,  // 5 x NCOLP
                             float* __restrict__ C) {         // 48 x NCOLP
    const int gtid = blockIdx.x * blockDim.x + threadIdx.x;
    const int wave = gtid >> 5;
    const int lane = gtid & 31;
    const int ntiles = NCOLP / 16;           // 2042
    if (wave >= 3 * ntiles) return;          // uniform per-wave exit

    const int mt = wave / ntiles;
    const int nt = wave % ntiles;
    const int m0 = mt * 16, n0 = nt * 16;
    const int lm = lane & 15;
    const int lh = lane >> 4;
    const int kb = lh * 2;
    const int m  = m0 + lm;
    const int n  = n0 + lm;

    v2f a0, a1, b0, b1;
    a0.x = Spad[m * 5 + kb];
    a0.y = Spad[m * 5 + kb + 1];
    const float s4 = Spad[m * 5 + 4];
    a1.x = (lh == 0) ? s4 : 0.0f;            // k=4 in slot 0 only
    a1.y = 0.0f;
    b0.x = Tpad[kb * NCOLP + n];
    b0.y = Tpad[(kb + 1) * NCOLP + n];
    const float t4 = Tpad[4 * NCOLP + n];
    b1.x = (lh == 0) ? t4 : 0.0f;
    b1.y = 0.0f;

    v8f acc = {};
    acc = __builtin_amdgcn_wmma_f32_16x16x4_f32(false, a0, false, b0,
                                                (short)0, acc, false, false);
    acc = __builtin_amdgcn_wmma_f32_16x16x4_f32(false, a1, false, b1,
                                                (short)0, acc, false, false);

#pragma unroll
    for (int v = 0; v < 8; ++v) {
        C[(m0 + v + lh * 8) * NCOLP + n] = acc[v];   // unconditional stores
    }
}

// ---------------- k5: d3lut = sum_n w[n] * cube_to_lut(cubes)[n] ------------
// cubes[n,ch,d2,d3,d4] = cubes2[d2, (n*3+ch)*1089 + d3*33 + d4]
//   ch=0 (r): lut[a,b,c] = cube[c,a,b];  ch=1 (g): cube[b,a,c];  ch=2: cube[a,b,c]
__global__ void k_d3lut(const float* __restrict__ cubes2,   // stride NCOLP
                        const float* __restrict__ weights,
                        float* __restrict__ d3lut) {
    const int i = blockIdx.x * blockDim.x + threadIdx.x;
    if (i >= 3 * CUBE3) return;
    const int ch = i / CUBE3;
    const int r  = i % CUBE3;
    const int a  = r / PLANE2;
    const int s2 = r % PLANE2;
    const int b  = s2 / DIMV;
    const int c  = s2 % DIMV;

    int d2, off;
    if (ch == 0)      { d2 = c; off = a * DIMV + b; }
    else if (ch == 1) { d2 = b; off = a * DIMV + c; }
    else              { d2 = a; off = b * DIMV + c; }

    const float* base = cubes2 + (size_t)d2 * NCOLP + ch * PLANE2 + off;
    float acc = 0.0f;
#pragma unroll
    for (int n = 0; n < NUMV; ++n)
        acc = fmaf(weights[n], base[(size_t)n * 3 * PLANE2], acc);
    d3lut[i] = acc;
}

// ---------------- k6: trilinear interpolation (4 px / thread) ---------------
// The two r-corners are adjacent in memory -> load as one {lo,hi} pair
// (align-4, size-8 -> compiler can merge into global_load_b64).
// 12 gathers/pixel instead of 24.
__global__ void k_interp(const float* __restrict__ img,
                         const float* __restrict__ L,    // 3*33^3
                         float* __restrict__ out) {
    const int q = blockIdx.x * blockDim.x + threadIdx.x;  // quad index
    const int NQ = HW_ORG / 4;
    if (q >= NQ) return;

    const float4* in4 = (const float4*)img;
    const float4 rv = in4[q];
    const float4 gv = in4[q + NQ];
    const float4 bv = in4[q + 2 * NQ];

    const float invbin = (float)(32.0 / 1.000001);   // 1/binsize

    float ri[4] = {rv.x, rv.y, rv.z, rv.w};
    float gi[4] = {gv.x, gv.y, gv.z, gv.w};
    float bi[4] = {bv.x, bv.y, bv.z, bv.w};
    float ro[4], go[4], bo[4];

#pragma unroll
    for (int t = 0; t < 4; ++t) {
        const float rf = ri[t] * invbin;
        const float gf = gi[t] * invbin;
        const float bf = bi[t] * invbin;
        const float rfl = floorf(rf), gfl = floorf(gf), bfl = floorf(bf);
        const int rid = iclamp((int)rfl, 0, DIMV - 2);
        const int gid = iclamp((int)gfl, 0, DIMV - 2);
        const int bid = iclamp((int)bfl, 0, DIMV - 2);
        const float rd = rf - rfl, gd = gf - gfl, bd = bf - bfl;
        const float rl = 1.0f - rd;
        const float gw[2] = {1.0f - gd, gd};
        const float bw[2] = {1.0f - bd, bd};
        const int base = bid * PLANE2 + gid * DIMV + rid;

        float o0 = 0.0f, o1 = 0.0f, o2 = 0.0f;
#pragma unroll
        for (int db = 0; db < 2; ++db)
#pragma unroll
            for (int dg = 0; dg < 2; ++dg) {
                const float wgb = gw[dg] * bw[db];
                const int idx = base + db * PLANE2 + dg * DIMV;
                const fpair p0 = *(const fpair*)(L + idx);
                const fpair p1 = *(const fpair*)(L + CUBE3 + idx);
                const fpair p2 = *(const fpair*)(L + 2 * CUBE3 + idx);
                o0 = fmaf(wgb, fmaf(p0.lo, rl, p0.hi * rd), o0);
                o1 = fmaf(wgb, fmaf(p1.lo, rl, p1.hi * rd), o1);
                o2 = fmaf(wgb, fmaf(p2.lo, rl, p2.hi * rd), o2);
            }
        ro[t] = o0 + ri[t];
        go[t] = o1 + gi[t];
        bo[t] = o2 + bi[t];
    }

    float4* out4 = (float4*)out;
    out4[q]          = make_float4(ro[0], ro[1], ro[2], ro[3]);
    out4[q + NQ]     = make_float4(go[0], go[1], go[2], go[3]);
    out4[q + 2 * NQ] = make_float4(bo[0], bo[1], bo[2], bo[3]);
}

// ---------------------------------------------------------------------------
extern "C" void kernel_launch(void* const* d_in, const int* in_sizes, int n_in,
                              void* d_out, int out_size, void* d_ws, size_t ws_size,
                              hipStream_t stream) {
    const int*   img_msb = (const int*)  d_in[0];
    const int*   img_lsb = (const int*)  d_in[1];
    const float* img_org = (const float*)d_in[2];
    const float* fmsb    = (const float*)d_in[3];
    const float* flsb    = (const float*)d_in[4];
    const float* lut_cat = (const float*)d_in[5];
    const float* s_layers= (const float*)d_in[6];
    const float* w_layers= (const float*)d_in[7];
    const float* luts    = (const float*)d_in[8];

    float* ws       = (float*)d_ws;
    float* partials = ws + WS_PARTIALS;
    float* weights  = ws + WS_WEIGHTS;
    float* Spad     = ws + WS_SPAD;
    float* Lpad     = ws + WS_LPAD;
    float* Wpad     = ws + WS_WPAD;
    float* Tpad     = ws + WS_TPAD;
    float* cubes2   = ws + WS_CUBES2;

    float* out_img  = (float*)d_out;                       // 3*2160*3840
    float* d3lut    = (float*)d_out + (size_t)3 * HW_ORG;  // 3*33^3

    // k1: 256 blocks x 256 threads covers 65536 sample pixels
    k_feat_partial<<<256, 256, 0, stream>>>(img_msb, img_lsb, fmsb, flsb, partials);
    // k2: single-block final reduction + weight selection
    k_weights<<<1, 256, 0, stream>>>(partials, lut_cat, weights);
    // kp: zero-pad WMMA operands
    k_pad<<<(13248 + 255) / 256, 256, 0, stream>>>(s_layers, luts, w_layers,
                                                   Spad, Lpad, Wpad);
    // k3: 10*69 = 690 tiles, one wave each -> WMMA
    k_lutsw_wmma<<<(690 * 32 + 255) / 256, 256, 0, stream>>>(Lpad, Wpad, Tpad);
    // k4: 3*2042 = 6126 tiles, one wave each -> WMMA (branch-free)
    k_cubes_wmma<<<(6126 * 32 + 255) / 256, 256, 0, stream>>>(Spad, Tpad, cubes2);
    // k5: 3*33^3 outputs -> tail of d_out
    k_d3lut<<<(3 * CUBE3 + 255) / 256, 256, 0, stream>>>(cubes2, weights, d3lut);
    // k6: 8294400/4 quads
    k_interp<<<(HW_ORG / 4 + 255) / 256, 256, 0, stream>>>(img_org, d3lut, out_img);
}